// ReachabilityClassifierTransformer_13065290514897
// MI455X (gfx1250) — compile-verified
//
#include <hip/hip_runtime.h>
#include <hip/hip_bf16.h>
#include <math.h>

typedef __attribute__((ext_vector_type(2))) float v2f;
typedef __attribute__((ext_vector_type(8))) float v8f;

#define LN_EPS 1e-5f
static constexpr int ROWS = 16 * 512;   // B*S = 8192

// ---------------------------------------------------------------------------
// Tiled fp32 WMMA GEMM:  C = alpha * (A @ op(B)) + bias + Res, optional ReLU
//   BT=true : B is (N,K) row-major (PyTorch weight, computes A @ B^T)
//   BT=false: B is (K,N) with row stride ldb
// Batched over blockIdx.z with zb=z/zdiv, zh=z%zdiv pointer offsets
// (used to walk (batch, head) sub-matrices for attention).
// fp32 WMMA 16x16x4 fragments per CDNA5 ISA 7.12.2:
//   A 16x4 : lanes0-15 {K=k,k+1}, lanes16-31 {K=k+2,k+3}, M = lane%16
//   B 4x16 : v0 = rows k (lanes0-15) / k+2 (lanes16-31), v1 = k+1 / k+3
//   C/D    : vgpr r -> row r (lanes0-15) / r+8 (lanes16-31), N = lane%16
// ---------------------------------------------------------------------------
template <int BN, bool BT>
__global__ __launch_bounds__(256) void k_gemm(
    const float* A, int lda,
    const float* Bm, int ldb,
    const float* bias,
    const float* Res,
    float* C, int ldc,
    int M, int N, int K, float alpha, int relu,
    int zdiv, long long sAo, long long sAi,
    long long sBo, long long sBi, long long sCo, long long sCi)
{
    constexpr int NS = BN / 32;       // 16-wide sub-tiles per wave
    __shared__ float As[64][34];
    __shared__ float Bs[BN][34];

    const int tid  = threadIdx.x;
    const int lane = tid & 31;
    const int wave = tid >> 5;
    const int wr   = wave & 3;        // wave row tile (16 rows)
    const int wc   = wave >> 2;       // wave col half (BN/2 cols)
    const int m16  = lane & 15;
    const int half = lane >> 4;

    const int zb = blockIdx.z / zdiv;
    const int zh = blockIdx.z % zdiv;
    A  += (long long)zb * sAo + (long long)zh * sAi;
    Bm += (long long)zb * sBo + (long long)zh * sBi;
    C  += (long long)zb * sCo + (long long)zh * sCi;
    if (Res) Res += (long long)zb * sCo + (long long)zh * sCi;

    const int row0 = blockIdx.y * 64;
    const int col0 = blockIdx.x * BN;

    v8f acc[NS];
    for (int s = 0; s < NS; ++s)
        for (int i = 0; i < 8; ++i) acc[s][i] = 0.0f;

    for (int k0 = 0; k0 < K; k0 += 32) {
        // ---- stage A tile (64 x 32), coalesced float4 ----
#pragma unroll
        for (int r = 0; r < 2; ++r) {
            int m  = r * 32 + (tid >> 3);
            int k4 = (tid & 7) * 4;
            int gm = row0 + m;
            float4 v = make_float4(0.f, 0.f, 0.f, 0.f);
            if (gm < M) v = *(const float4*)(A + (size_t)gm * lda + k0 + k4);
            As[m][k4 + 0] = v.x; As[m][k4 + 1] = v.y;
            As[m][k4 + 2] = v.z; As[m][k4 + 3] = v.w;
        }
        // ---- stage B tile (BN x 32) ----
        if (BT) {
#pragma unroll
            for (int r = 0; r < BN / 32; ++r) {
                int n  = r * 32 + (tid >> 3);
                int k4 = (tid & 7) * 4;
                int gn = col0 + n;
                float4 v = make_float4(0.f, 0.f, 0.f, 0.f);
                if (gn < N) v = *(const float4*)(Bm + (size_t)gn * ldb + k0 + k4);
                Bs[n][k4 + 0] = v.x; Bs[n][k4 + 1] = v.y;
                Bs[n][k4 + 2] = v.z; Bs[n][k4 + 3] = v.w;
            }
        } else {
            constexpr int TPR = BN / 4;            // threads per K-row
#pragma unroll
            for (int r = 0; r < BN / 32; ++r) {
                int kk = r * (256 / TPR) + tid / TPR;
                int n4 = (tid % TPR) * 4;
                float4 v = make_float4(0.f, 0.f, 0.f, 0.f);
                if (col0 + n4 < N)
                    v = *(const float4*)(Bm + (size_t)(k0 + kk) * ldb + col0 + n4);
                Bs[n4 + 0][kk] = v.x; Bs[n4 + 1][kk] = v.y;
                Bs[n4 + 2][kk] = v.z; Bs[n4 + 3][kk] = v.w;
            }
        }
        __syncthreads();

        // ---- 8 x WMMA f32 16x16x4 per sub-tile ----
#pragma unroll
        for (int kk = 0; kk < 32; kk += 4) {
            v2f a;
            a.x = As[wr * 16 + m16][kk + half * 2 + 0];
            a.y = As[wr * 16 + m16][kk + half * 2 + 1];
#pragma unroll
            for (int s = 0; s < NS; ++s) {
                v2f b;
                b.x = Bs[wc * (BN / 2) + s * 16 + m16][kk + half * 2 + 0];
                b.y = Bs[wc * (BN / 2) + s * 16 + m16][kk + half * 2 + 1];
                acc[s] = __builtin_amdgcn_wmma_f32_16x16x4_f32(
                    false, a, false, b, (short)0, acc[s], false, false);
            }
        }
        __syncthreads();
    }

    // ---- epilogue: alpha, bias, residual, relu ----
#pragma unroll
    for (int s = 0; s < NS; ++s) {
        int gc = col0 + wc * (BN / 2) + s * 16 + m16;
        if (gc >= N) continue;
        float bv = bias ? bias[gc] : 0.0f;
#pragma unroll
        for (int r = 0; r < 8; ++r) {
            int gr = row0 + wr * 16 + half * 8 + r;
            if (gr >= M) continue;
            float v = acc[s][r] * alpha + bv;
            if (Res) v += Res[(size_t)gr * ldc + gc];
            if (relu) v = fmaxf(v, 0.0f);
            C[(size_t)gr * ldc + gc] = v;
        }
    }
}

// ---------------------------------------------------------------------------
// LayerNorm over width 512, one block per row
// ---------------------------------------------------------------------------
__global__ __launch_bounds__(256) void k_layernorm(
    const float* __restrict__ x, const float* __restrict__ g,
    const float* __restrict__ b, float* __restrict__ y)
{
    __shared__ float red[16];
    const int tid = threadIdx.x, lane = tid & 31, wave = tid >> 5;
    const size_t r = blockIdx.x;
    float2 v = ((const float2*)(x + r * 512))[tid];
    float s  = v.x + v.y;
    float sq = v.x * v.x + v.y * v.y;
    for (int o = 16; o; o >>= 1) { s += __shfl_xor(s, o, 32); sq += __shfl_xor(sq, o, 32); }
    if (lane == 0) { red[wave] = s; red[8 + wave] = sq; }
    __syncthreads();
    if (tid == 0) {
        float ts = 0.f, tq = 0.f;
        for (int i = 0; i < 8; ++i) { ts += red[i]; tq += red[8 + i]; }
        red[0] = ts; red[8] = tq;
    }
    __syncthreads();
    const float mean = red[0] * (1.0f / 512.0f);
    const float var  = red[8] * (1.0f / 512.0f) - mean * mean;
    const float rstd = rsqrtf(var + LN_EPS);
    float2 gg = ((const float2*)g)[tid];
    float2 bb = ((const float2*)b)[tid];
    float2 o;
    o.x = (v.x - mean) * rstd * gg.x + bb.x;
    o.y = (v.y - mean) * rstd * gg.y + bb.y;
    ((float2*)(y + r * 512))[tid] = o;
}

// ---------------------------------------------------------------------------
// Softmax over width 512, in place, one block per row
// ---------------------------------------------------------------------------
__global__ __launch_bounds__(256) void k_softmax(float* __restrict__ s_)
{
    __shared__ float red[16];
    const int tid = threadIdx.x, lane = tid & 31, wave = tid >> 5;
    float2* row = (float2*)(s_ + (size_t)blockIdx.x * 512);
    float2 v = row[tid];
    float mx = fmaxf(v.x, v.y);
    for (int o = 16; o; o >>= 1) mx = fmaxf(mx, __shfl_xor(mx, o, 32));
    if (lane == 0) red[wave] = mx;
    __syncthreads();
    if (tid == 0) { float m = red[0]; for (int i = 1; i < 8; ++i) m = fmaxf(m, red[i]); red[0] = m; }
    __syncthreads();
    mx = red[0];
    float ex = __expf(v.x - mx), ey = __expf(v.y - mx);
    float sum = ex + ey;
    for (int o = 16; o; o >>= 1) sum += __shfl_xor(sum, o, 32);
    if (lane == 0) red[8 + wave] = sum;
    __syncthreads();
    if (tid == 0) { float t = 0.f; for (int i = 0; i < 8; ++i) t += red[8 + i]; red[8] = t; }
    __syncthreads();
    const float inv = 1.0f / red[8];
    row[tid] = make_float2(ex * inv, ey * inv);
}

// ---------------------------------------------------------------------------
// RoPE: reads q,k slices of qkv (row stride 1536), writes rearranged qr/kr
// qr[:,j]     = q[2j]*cos - q[2j+1]*sin
// qr[:,256+j] = q[2j]*sin + q[2j+1]*cos        (reference concat layout)
// ---------------------------------------------------------------------------
__global__ __launch_bounds__(256) void k_rope(const float* __restrict__ qkv,
                                              float* __restrict__ qr,
                                              float* __restrict__ kr)
{
    const int row = blockIdx.x;
    const int j   = threadIdx.x;            // 0..255
    const int s   = row & 511;              // sequence position
    const float e   = (float)(2 * j) * (1.0f / 512.0f);
    const float ang = (float)s * __powf(10000.0f, -e);
    float sn, cs;
    __sincosf(ang, &sn, &cs);
    const float* base = qkv + (size_t)row * 1536;
    const float q1 = base[2 * j],       q2 = base[2 * j + 1];
    const float k1 = base[512 + 2 * j], k2 = base[512 + 2 * j + 1];
    const size_t o = (size_t)row * 512;
    qr[o + j]       = q1 * cs - q2 * sn;
    qr[o + 256 + j] = q1 * sn + q2 * cs;
    kr[o + j]       = k1 * cs - k2 * sn;
    kr[o + 256 + j] = k1 * sn + k2 * cs;
}

// Small-K input projections (K = 3 or 9), always ReLU, N = 512
__global__ void k_proj_small(const float* __restrict__ in, int K,
                             const float* __restrict__ W,
                             const float* __restrict__ bias,
                             float* __restrict__ out, int rows)
{
    const int idx = blockIdx.x * blockDim.x + threadIdx.x;
    if (idx >= rows * 512) return;
    const int r = idx >> 9, n = idx & 511;
    const float* ip = in + (size_t)r * K;
    const float* wp = W + (size_t)n * K;
    float acc = bias[n];
    for (int k = 0; k < K; ++k) acc += ip[k] * wp[k];
    out[idx] = fmaxf(acc, 0.0f);
}

// Decoder single-query attention scores: sc[(b*8+h)*512+k] = <q_bh, K_bk_h>/8
__global__ void k_dec_scores(const float* __restrict__ q,
                             const float* __restrict__ kmem,
                             float* __restrict__ sc)
{
    const int idx = blockIdx.x * blockDim.x + threadIdx.x;
    if (idx >= 16 * 8 * 512) return;
    const int b = idx >> 12, h = (idx >> 9) & 7, k = idx & 511;
    const float* qp = q + b * 512 + h * 64;
    const float* kp = kmem + ((size_t)(b * 512 + k)) * 512 + h * 64;
    float acc = 0.f;
#pragma unroll
    for (int j = 0; j < 64; ++j) acc += qp[j] * kp[j];
    sc[idx] = acc * 0.125f;
}

// Decoder attention output: o[b, d] = sum_k P[b, d/64, k] * V[b, k, d]
__global__ void k_dec_attnout(const float* __restrict__ p,
                              const float* __restrict__ vmem,
                              float* __restrict__ o)
{
    const int idx = blockIdx.x * blockDim.x + threadIdx.x;
    if (idx >= 16 * 512) return;
    const int b = idx >> 9, d = idx & 511, h = d >> 6;
    const float* pp = p + ((size_t)(b * 8 + h)) * 512;
    const float* vp = vmem + (size_t)b * 512 * 512 + d;
    float acc = 0.f;
    for (int k = 0; k < 512; ++k) acc += pp[k] * vp[(size_t)k * 512];
    o[idx] = acc;
}

// Final head: sigmoid(hg @ head_w^T + bias), 16 outputs
__global__ void k_head(const float* __restrict__ hg, const float* __restrict__ w,
                       const float* __restrict__ b0, float* __restrict__ out)
{
    const int t = threadIdx.x;
    if (t < 16) {
        const float* row = hg + (size_t)t * 512;
        float acc = b0[0];
        for (int k = 0; k < 512; ++k) acc += row[k] * w[k];
        out[t] = 1.0f / (1.0f + __expf(-acc));
    }
}

// ---------------------------------------------------------------------------
// Host-side launch helpers
// ---------------------------------------------------------------------------
static void gemmBT128(hipStream_t st, const float* A, int lda, const float* W, int ldb,
                      const float* bias, const float* Res, float* C, int ldc,
                      int M, int N, int K, float alpha, int relu,
                      int zc, int zdiv, long long sAo, long long sAi,
                      long long sBo, long long sBi, long long sCo, long long sCi)
{
    dim3 g((unsigned)((N + 127) / 128), (unsigned)((M + 63) / 64), (unsigned)zc);
    hipLaunchKernelGGL((k_gemm<128, true>), g, dim3(256), 0, st,
                       A, lda, W, ldb, bias, Res, C, ldc, M, N, K, alpha, relu,
                       zdiv, sAo, sAi, sBo, sBi, sCo, sCi);
}
static void gemm(hipStream_t st, const float* A, int lda, const float* W, int ldb,
                 const float* bias, const float* Res, float* C, int ldc,
                 int M, int N, int K, float alpha, int relu)
{
    gemmBT128(st, A, lda, W, ldb, bias, Res, C, ldc, M, N, K, alpha, relu,
              1, 1, 0, 0, 0, 0, 0, 0);
}
static void gemmNT64(hipStream_t st, const float* A, int lda, const float* Bm, int ldb,
                     const float* bias, const float* Res, float* C, int ldc,
                     int M, int N, int K, float alpha, int relu,
                     int zc, int zdiv, long long sAo, long long sAi,
                     long long sBo, long long sBi, long long sCo, long long sCi)
{
    dim3 g((unsigned)((N + 63) / 64), (unsigned)((M + 63) / 64), (unsigned)zc);
    hipLaunchKernelGGL((k_gemm<64, false>), g, dim3(256), 0, st,
                       A, lda, Bm, ldb, bias, Res, C, ldc, M, N, K, alpha, relu,
                       zdiv, sAo, sAi, sBo, sBi, sCo, sCi);
}

extern "C" void kernel_launch(void* const* d_in, const int* in_sizes, int n_in,
                              void* d_out, int out_size, void* d_ws, size_t ws_size,
                              hipStream_t stream)
{
    (void)in_sizes; (void)n_in; (void)out_size; (void)ws_size;

    const float* pose         = (const float*)d_in[0];
    const float* morph        = (const float*)d_in[1];
    const float* morph_proj_w = (const float*)d_in[2];
    const float* morph_proj_b = (const float*)d_in[3];
    const float* enc_in_w     = (const float*)d_in[4];
    const float* enc_in_b     = (const float*)d_in[5];
    const float* enc_out_w    = (const float*)d_in[6];
    const float* enc_out_b    = (const float*)d_in[7];
    const float* enc_n1_g     = (const float*)d_in[8];
    const float* enc_n1_b     = (const float*)d_in[9];
    const float* enc_n2_g     = (const float*)d_in[10];
    const float* enc_n2_b     = (const float*)d_in[11];
    const float* enc_l1_w     = (const float*)d_in[12];
    const float* enc_l1_b     = (const float*)d_in[13];
    const float* enc_l2_w     = (const float*)d_in[14];
    const float* enc_l2_b     = (const float*)d_in[15];
    const float* enc_final_g  = (const float*)d_in[16];
    const float* enc_final_b  = (const float*)d_in[17];
    const float* pose_proj_w  = (const float*)d_in[18];
    const float* pose_proj_b  = (const float*)d_in[19];
    const float* dec_in_w     = (const float*)d_in[20];
    const float* dec_in_b     = (const float*)d_in[21];
    const float* dec_out_w    = (const float*)d_in[22];
    const float* dec_out_b    = (const float*)d_in[23];
    const float* dec_n1_g     = (const float*)d_in[24];
    const float* dec_n1_b     = (const float*)d_in[25];
    const float* dec_n2_g     = (const float*)d_in[26];
    const float* dec_n2_b     = (const float*)d_in[27];
    const float* dec_m1_w     = (const float*)d_in[28];
    const float* dec_m1_b     = (const float*)d_in[29];
    const float* dec_m2_w     = (const float*)d_in[30];
    const float* dec_m2_b     = (const float*)d_in[31];
    const float* head_g       = (const float*)d_in[32];
    const float* head_b       = (const float*)d_in[33];
    const float* head_w       = (const float*)d_in[34];
    const float* head_bias    = (const float*)d_in[35];

    // ---- workspace layout (floats); lifetimes allow heavy aliasing ----
    float* ws = (float*)d_ws;
    const size_t MF = 1024 * 1024;
    float* X    = ws + 0 * MF;    // 4M  residual stream (8192x512)
    float* Hb   = ws + 4 * MF;    // 4M  LN output / decoder small buffers
    float* QKV  = ws + 8 * MF;    // 12M qkv (8192x1536)
    float* QR   = ws + 20 * MF;   // 4M  roped q
    float* KR   = ws + 24 * MF;   // 4M  roped k
    float* Q2   = ws + 28 * MF;   // 4M  second q projection
    float* K2   = ws + 32 * MF;   // 4M  second k projection
    float* V2   = ws + 36 * MF;   // 4M  v projection
    float* SC   = ws + 40 * MF;   // 8.4M scores chunk (4 batches x 8 heads x 512 x 512)
    float* AO   = ws + 8 * MF;    // 4M  attention out  (aliases dead QKV)
    float* FFNH = ws + 8 * MF;    // 16M ffn hidden     (aliases dead QKV/QR)
    // decoder (after encoder; QKV/QR regions repurposed)
    float* MENC = QR;             // encoded memory (8192x512)
    float* KM   = ws + 8 * MF;    // 4M decoder K memory
    float* VM   = ws + 12 * MF;   // 4M decoder V memory
    float* P    = Hb;             // 16x512
    float* Y    = Hb + 8192;
    float* QLN  = Hb + 16384;
    float* QD   = Hb + 24576;
    float* OD   = Hb + 32768;
    float* DFF  = Hb + 40960;     // 16x2048
    float* DSC  = Hb + 81920;     // 16x8x512

    const long long HB = 512LL * 512LL;        // per-(b,h) score block / per-b token block

    // ================= encoder =================
    k_proj_small<<<(ROWS * 512 + 255) / 256, 256, 0, stream>>>(
        morph, 3, morph_proj_w, morph_proj_b, X, ROWS);

    for (int i = 0; i < 6; ++i) {
        const float* Wi = enc_in_w + (size_t)i * 1536 * 512;
        const float* bi = enc_in_b + (size_t)i * 1536;

        k_layernorm<<<ROWS, 256, 0, stream>>>(X, enc_n1_g + i * 512, enc_n1_b + i * 512, Hb);
        gemm(stream, Hb, 512, Wi, 512, bi, nullptr, QKV, 1536, ROWS, 1536, 512, 1.f, 0);
        k_rope<<<ROWS, 256, 0, stream>>>(QKV, QR, KR);

        // second projection inside _mha (reference quirk)
        gemm(stream, QR, 512, Wi, 512, bi, nullptr, Q2, 512, ROWS, 512, 512, 1.f, 0);
        gemm(stream, KR, 512, Wi + 512 * 512, 512, bi + 512, nullptr, K2, 512, ROWS, 512, 512, 1.f, 0);
        gemm(stream, QKV + 1024, 1536, Wi + 1024 * 512, 512, bi + 1024, nullptr, V2, 512, ROWS, 512, 512, 1.f, 0);

        // attention, 4 batches per chunk (32 (b,h) blocks in grid.z)
        for (int c = 0; c < 4; ++c) {
            const long long base = (long long)c * 4 * HB;
            gemmBT128(stream, Q2 + base, 512, K2 + base, 512, nullptr, nullptr, SC, 512,
                      512, 512, 64, 0.125f, 0,
                      32, 8, HB, 64, HB, 64, 8 * HB, HB);
            k_softmax<<<4 * 8 * 512, 256, 0, stream>>>(SC);
            gemmNT64(stream, SC, 512, V2 + base, 512, nullptr, nullptr, AO + base, 512,
                     512, 64, 512, 1.0f, 0,
                     32, 8, 8 * HB, HB, HB, 64, HB, 64);
        }

        // output projection + residual
        gemm(stream, AO, 512, enc_out_w + (size_t)i * 512 * 512, 512,
             enc_out_b + i * 512, X, X, 512, ROWS, 512, 512, 1.f, 0);

        // FFN
        k_layernorm<<<ROWS, 256, 0, stream>>>(X, enc_n2_g + i * 512, enc_n2_b + i * 512, Hb);
        gemm(stream, Hb, 512, enc_l1_w + (size_t)i * 2048 * 512, 512,
             enc_l1_b + i * 2048, nullptr, FFNH, 2048, ROWS, 2048, 512, 1.f, 1);
        gemm(stream, FFNH, 2048, enc_l2_w + (size_t)i * 512 * 2048, 2048,
             enc_l2_b + i * 512, X, X, 512, ROWS, 512, 2048, 1.f, 0);
    }
    k_layernorm<<<ROWS, 256, 0, stream>>>(X, enc_final_g, enc_final_b, MENC);

    // ================= decoder =================
    k_proj_small<<<(16 * 512 + 255) / 256, 256, 0, stream>>>(
        pose, 9, pose_proj_w, pose_proj_b, P, 16);

    for (int i = 0; i < 4; ++i) {
        const float* Wi = dec_in_w + (size_t)i * 1536 * 512;
        const float* bi = dec_in_b + (size_t)i * 1536;

        k_layernorm<<<16, 256, 0, stream>>>(P, dec_n1_g + i * 512, dec_n1_b + i * 512, QLN);
        gemm(stream, QLN, 512, Wi, 512, bi, nullptr, QD, 512, 16, 512, 512, 1.f, 0);
        gemm(stream, MENC, 512, Wi + 512 * 512, 512, bi + 512, nullptr, KM, 512, ROWS, 512, 512, 1.f, 0);
        gemm(stream, MENC, 512, Wi + 1024 * 512, 512, bi + 1024, nullptr, VM, 512, ROWS, 512, 512, 1.f, 0);

        k_dec_scores<<<(16 * 8 * 512 + 255) / 256, 256, 0, stream>>>(QD, KM, DSC);
        k_softmax<<<16 * 8, 256, 0, stream>>>(DSC);
        k_dec_attnout<<<(16 * 512 + 255) / 256, 256, 0, stream>>>(DSC, VM, OD);

        gemm(stream, OD, 512, dec_out_w + (size_t)i * 512 * 512, 512,
             dec_out_b + i * 512, P, Y, 512, 16, 512, 512, 1.f, 0);

        k_layernorm<<<16, 256, 0, stream>>>(Y, dec_n2_g + i * 512, dec_n2_b + i * 512, QLN);
        gemm(stream, QLN, 512, dec_m1_w + (size_t)i * 2048 * 512, 512,
             dec_m1_b + i * 2048, nullptr, DFF, 2048, 16, 2048, 512, 1.f, 1);
        gemm(stream, DFF, 2048, dec_m2_w + (size_t)i * 512 * 2048, 2048,
             dec_m2_b + i * 512, Y, P, 512, 16, 512, 2048, 1.f, 0);
    }

    k_layernorm<<<16, 256, 0, stream>>>(P, head_g, head_b, QLN);
    k_head<<<1, 256, 0, stream>>>(QLN, head_w, head_bias, (float*)d_out);
}